// rLayer_31318901523048
// MI455X (gfx1250) — compile-verified
//
#include <hip/hip_runtime.h>
#include <math.h>

// Problem constants (match reference)
#define MDIM  16384
#define NDIM  65536
#define NNZC  4194304
#define BATCH 16

typedef __attribute__((ext_vector_type(2))) float v2f;
typedef __attribute__((ext_vector_type(8))) float v8f;

// ---------------------------------------------------------------------------
// K0: zero the accumulation region (t1 | t2T | norm2 are contiguous in ws)
// ---------------------------------------------------------------------------
__global__ void k_zero(float* __restrict__ p, int n) {
  int i = blockIdx.x * blockDim.x + threadIdx.x;
  if (i < n) p[i] = 0.0f;
}

// ---------------------------------------------------------------------------
// K1: zuT[n][b] = z[b][n]*u[b][n];  uT[n][b] = u[b][n]
// One thread per column n: reads of z/u are coalesced across lanes per b,
// writes are 64B-contiguous float4 bursts per thread.
// ---------------------------------------------------------------------------
__global__ void k_pack(const float* __restrict__ z, const float* __restrict__ u,
                       float* __restrict__ zuT, float* __restrict__ uT) {
  int n = blockIdx.x * blockDim.x + threadIdx.x;   // 0..NDIM-1
  float zv[BATCH], uv[BATCH];
#pragma unroll
  for (int b = 0; b < BATCH; ++b) {
    float uu = u[(size_t)b * NDIM + n];
    uv[b] = uu;
    zv[b] = z[(size_t)b * NDIM + n] * uu;
  }
  float4* z4 = (float4*)(zuT + (size_t)n * BATCH);
  float4* u4 = (float4*)(uT  + (size_t)n * BATCH);
#pragma unroll
  for (int j = 0; j < 4; ++j) {
    z4[j] = make_float4(zv[4*j], zv[4*j+1], zv[4*j+2], zv[4*j+3]);
    u4[j] = make_float4(uv[4*j], uv[4*j+1], uv[4*j+2], uv[4*j+3]);
  }
}

// ---------------------------------------------------------------------------
// K2/K4: COO SpMM, 16 lanes per nonzero. Gather 64B line, atomic 64B line.
// dst[scatter[k]][b] += vals[k] * src[gather[k]][b]
// ---------------------------------------------------------------------------
__global__ void k_spmm(const float* __restrict__ vals,
                       const int* __restrict__ gather_idx,
                       const int* __restrict__ scatter_idx,
                       const float* __restrict__ src,   // [*, 16] transposed
                       float* __restrict__ dst) {       // [*, 16] transposed
  unsigned tid = blockIdx.x * blockDim.x + threadIdx.x;
  unsigned b = tid & 15u;
  unsigned k = tid >> 4;               // < NNZC by grid construction
  if (b == 0u && k + 4096u < NNZC) {   // run-ahead on the streaming COO arrays
    __builtin_prefetch(&vals[k + 4096u], 0, 0);        // global_prefetch_b8
    __builtin_prefetch(&gather_idx[k + 4096u], 0, 0);
    __builtin_prefetch(&scatter_idx[k + 4096u], 0, 0);
  }
  float v = vals[k];
  int g = gather_idx[k];
  int s = scatter_idx[k];
  float contrib = v * src[(size_t)g * BATCH + b];
  atomicAdd(&dst[(size_t)s * BATCH + b], contrib);
}

// ---------------------------------------------------------------------------
// K3: r[m][b] = y[b][m] - t1[m][b]   (in place over t1)
// ---------------------------------------------------------------------------
__global__ void k_resid(const float* __restrict__ y, float* __restrict__ t1) {
  int m = blockIdx.x * blockDim.x + threadIdx.x;      // 0..MDIM-1
  float rv[BATCH];
  float4* t4 = (float4*)(t1 + (size_t)m * BATCH);
  float4 a0 = t4[0], a1 = t4[1], a2 = t4[2], a3 = t4[3];
  float tv[BATCH] = {a0.x,a0.y,a0.z,a0.w, a1.x,a1.y,a1.z,a1.w,
                     a2.x,a2.y,a2.z,a2.w, a3.x,a3.y,a3.z,a3.w};
#pragma unroll
  for (int b = 0; b < BATCH; ++b)
    rv[b] = y[(size_t)b * MDIM + m] - tv[b];
#pragma unroll
  for (int j = 0; j < 4; ++j)
    t4[j] = make_float4(rv[4*j], rv[4*j+1], rv[4*j+2], rv[4*j+3]);
}

// ---------------------------------------------------------------------------
// K5: per-batch squared L2 norm of v = u .* t2 via the matrix pipe.
// Gram trick: D += Vc * Vc^T with V_WMMA_F32_16X16X4_F32, Vc = 16x4 chunk of
// V (batch x n). Under the CDNA5 operand layouts (A 16x4: lanes0-15 K=0/1,
// lanes16-31 K=2/3; B 4x16 mirrored), the A and B per-lane registers of a
// symmetric product are identical, so one float2 feeds both operands.
// Diagonal of the 16x16 f32 accumulator = the 16 squared norms.
// Each wave covers 64 columns (16 WMMAs); 1024 waves cover N=65536.
// ---------------------------------------------------------------------------
__global__ void k_norm_wmma(const float* __restrict__ uT,
                            const float* __restrict__ t2T,
                            float* __restrict__ norm2) {
  unsigned tid  = blockIdx.x * blockDim.x + threadIdx.x;
  unsigned wave = tid >> 5;
  unsigned lane = tid & 31u;
  unsigned m  = lane & 15u;            // batch row
  unsigned kb = (lane >> 4) * 2u;      // K base: 0 for lanes0-15, 2 for 16-31
  unsigned n_start = wave * 64u;

  v8f c = {};
#pragma unroll
  for (int j = 0; j < 16; ++j) {
    size_t col = (size_t)n_start + (size_t)j * 4 + kb;
    size_t idx = col * BATCH + m;
    v2f a;
    a.x = uT[idx]         * t2T[idx];          // v[m][col]
    a.y = uT[idx + BATCH] * t2T[idx + BATCH];  // v[m][col+1]
    // D = A*B + C ; A == B (symmetric Gram), accumulate in f32
    c = __builtin_amdgcn_wmma_f32_16x16x4_f32(
        /*neg_a=*/false, a, /*neg_b=*/false, a,
        /*c_mod=*/(short)0, c, /*reuse_a=*/false, /*reuse_b=*/false);
  }
  // Diagonal extraction per C/D layout: (b,b) for b<8 at lane=b,vgpr=b;
  // for b>=8 at lane=b+16, vgpr=b-8.
  float d = 0.0f;
  int b = -1;
  if (lane < 8u)        { d = c[lane];        b = (int)lane; }
  else if (lane >= 24u) { d = c[lane - 24u];  b = (int)lane - 16; }
  if (b >= 0) atomicAdd(&norm2[b], d);
}

// ---------------------------------------------------------------------------
// K6: out[b][n] = z[b][n] - (eta / max(1, ||v_b||)) * u[b][n]*t2[b][n]
// ---------------------------------------------------------------------------
__global__ void k_update(const float* __restrict__ z,
                         const float* __restrict__ uT,
                         const float* __restrict__ t2T,
                         const float* __restrict__ norm2,
                         const float* __restrict__ eta,
                         float* __restrict__ out) {
  int n = blockIdx.x * blockDim.x + threadIdx.x;     // 0..NDIM-1
  float e = eta[0];
  const float4* u4 = (const float4*)(uT  + (size_t)n * BATCH);
  const float4* t4 = (const float4*)(t2T + (size_t)n * BATCH);
  float4 ua = u4[0], ub = u4[1], uc = u4[2], ud = u4[3];
  float4 ta = t4[0], tb = t4[1], tc = t4[2], td = t4[3];
  float uv[BATCH] = {ua.x,ua.y,ua.z,ua.w, ub.x,ub.y,ub.z,ub.w,
                     uc.x,uc.y,uc.z,uc.w, ud.x,ud.y,ud.z,ud.w};
  float tv[BATCH] = {ta.x,ta.y,ta.z,ta.w, tb.x,tb.y,tb.z,tb.w,
                     tc.x,tc.y,tc.z,tc.w, td.x,td.y,td.z,td.w};
#pragma unroll
  for (int b = 0; b < BATCH; ++b) {
    float coef = e / fmaxf(1.0f, sqrtf(norm2[b]));
    out[(size_t)b * NDIM + n] = z[(size_t)b * NDIM + n] - coef * uv[b] * tv[b];
  }
}

// ---------------------------------------------------------------------------
extern "C" void kernel_launch(void* const* d_in, const int* in_sizes, int n_in,
                              void* d_out, int out_size, void* d_ws, size_t ws_size,
                              hipStream_t stream) {
  const float* z      = (const float*)d_in[0];
  const float* u      = (const float*)d_in[1];
  const float* y      = (const float*)d_in[2];
  const float* A_vals = (const float*)d_in[3];
  const int*   A_rows = (const int*)d_in[4];
  const int*   A_cols = (const int*)d_in[5];
  const float* eta    = (const float*)d_in[6];
  float* out = (float*)d_out;

  // Workspace layout (floats), ~13.6 MB total:
  float* ws    = (float*)d_ws;
  float* zuT   = ws;                            // [NDIM][16]
  float* uT    = zuT + (size_t)NDIM * BATCH;    // [NDIM][16]
  float* t1    = uT  + (size_t)NDIM * BATCH;    // [MDIM][16]  (becomes r)
  float* t2T   = t1  + (size_t)MDIM * BATCH;    // [NDIM][16]
  float* norm2 = t2T + (size_t)NDIM * BATCH;    // [16]

  // Zero the accumulation region (t1 | t2T | norm2 contiguous)
  int zelems = MDIM * BATCH + NDIM * BATCH + BATCH;
  k_zero<<<(zelems + 255) / 256, 256, 0, stream>>>(t1, zelems);

  // zu and u transposed into [n][16] lines
  k_pack<<<NDIM / 256, 256, 0, stream>>>(z, u, zuT, uT);

  // SpMM1: t1[row][b] += val * zuT[col][b]
  unsigned spthreads = (unsigned)NNZC * 16u;
  k_spmm<<<spthreads / 256, 256, 0, stream>>>(A_vals, A_cols, A_rows, zuT, t1);

  // r = y - t1 (in place)
  k_resid<<<MDIM / 256, 256, 0, stream>>>(y, t1);

  // SpMM2: t2T[col][b] += val * r[row][b]
  k_spmm<<<spthreads / 256, 256, 0, stream>>>(A_vals, A_rows, A_cols, t1, t2T);

  // per-batch ||v||^2 on the matrix pipe (1024 waves, 64 cols each)
  k_norm_wmma<<<(NDIM / 64) * 32 / 256, 256, 0, stream>>>(uT, t2T, norm2);

  // final clamped gradient step
  k_update<<<NDIM / 256, 256, 0, stream>>>(z, uT, t2T, norm2, eta, out);
}